// PANCrafter_29274497089821
// MI455X (gfx1250) — compile-verified
//
#include <hip/hip_runtime.h>
#include <hip/hip_bf16.h>

typedef __attribute__((ext_vector_type(2))) float v2f;
typedef __attribute__((ext_vector_type(8))) float v8f;

#define HWSZ 16384   // 128*128
#define IMW 128

__device__ inline v8f wmma_f32(v2f a, v2f b, v8f c) {
  return __builtin_amdgcn_wmma_f32_16x16x4_f32(
      /*neg_a=*/false, a, /*neg_b=*/false, b,
      /*c_mod=*/(short)0, c, /*reuse_a=*/false, /*reuse_b=*/false);
}

// ---------------------------------------------------------------------------
// Kernel 0a: build 19 extra channels per batch:
//   xe[b][0..7] = cond = lpan*(1-s)+ms*s ; [8..15]=ms ; [16]=lpan ; [17]=pan ;
//   [18]=pan-lpan
// ---------------------------------------------------------------------------
__global__ __launch_bounds__(256) void build_xe(
    const float* __restrict__ ms, const float* __restrict__ lpan,
    const float* __restrict__ pan, const float* __restrict__ sv,
    float* __restrict__ xe)
{
  int idx = blockIdx.x * 256 + threadIdx.x;
  if (idx >= 2 * HWSZ) return;
  int b = idx / HWSZ, pix = idx % HWSZ;
  float s  = sv[b];
  float lp = lpan[(size_t)b * HWSZ + pix];
  float pn = pan [(size_t)b * HWSZ + pix];
  float* o = xe + (size_t)b * 19 * HWSZ + pix;
  #pragma unroll
  for (int c = 0; c < 8; ++c) {
    float m = ms[((size_t)b * 8 + c) * HWSZ + pix];
    o[(size_t)c       * HWSZ] = lp * (1.f - s) + m * s;
    o[(size_t)(8 + c) * HWSZ] = m;
  }
  o[(size_t)16 * HWSZ] = lp;
  o[(size_t)17 * HWSZ] = pn;
  o[(size_t)18 * HWSZ] = pn - lp;
}

// ---------------------------------------------------------------------------
// Kernel 0b: pack conv weights (O,I,3,3) -> [tap][cpair][oc][2], zero-padded
// so A fragments are single coalesced b64 loads with no predication.
// ---------------------------------------------------------------------------
__global__ __launch_bounds__(256) void pack_w(
    const float* __restrict__ w, float* __restrict__ dst,
    int cin, int cpad, int oc)
{
  int idx = blockIdx.x * 256 + threadIdx.x;
  int total = 9 * cpad * oc;
  if (idx >= total) return;
  int tap = idx / (cpad * oc);
  int rem = idx % (cpad * oc);
  int c   = rem / oc;
  int o   = rem % oc;
  float v = (c < cin) ? w[(size_t)o * cin * 9 + (size_t)c * 9 + tap] : 0.f;
  dst[(((size_t)tap * (cpad >> 1) + (c >> 1)) * oc + o) * 2 + (c & 1)] = v;
}

// Kernel 0c: pack both 1x1 projection weights -> [s][cpair][128][2]
__global__ __launch_bounds__(256) void pack_proj(
    const float* __restrict__ wpan, const float* __restrict__ wms,
    float* __restrict__ dst)
{
  int idx = blockIdx.x * 256 + threadIdx.x;
  if (idx >= 2 * 128 * 128) return;
  int s = idx >> 14;
  int rem = idx & 16383;
  int c = rem >> 7;
  int o = rem & 127;
  const float* w = s ? wms : wpan;
  dst[(size_t)s * 16384 + (((size_t)(c >> 1)) * 128 + o) * 2 + (c & 1)] =
      w[(size_t)o * 128 + c];
}

// ---------------------------------------------------------------------------
// Kernel 1: one branch of the front convs as an LDS-staged implicit GEMM.
// Block = one 16-pixel strip; 8 waves = 8 M-tiles (128 out channels).
// Inner loop per WMMA: 1 global_load_b64 (packed weights) + 1 ds_load_b64.
// ---------------------------------------------------------------------------
template<int CIN, int CPAD, int OC>
__global__ __launch_bounds__(256) void conv_front_t(
    const float* __restrict__ x,  const float* __restrict__ xe, int xebase,
    const float* __restrict__ wpk, const float* __restrict__ bias,
    int mOutBase, int mWBase, float oscale, float* __restrict__ out5)
{
  // patch[cpair][54 (+2 pad)][2] : 3 rows x 18 cols halo strip, zero padded
  __shared__ float patch[(CPAD / 2) * 56 * 2];

  int t = threadIdx.x;
  int nBase = blockIdx.x * 16;
  int b   = nBase / HWSZ;
  int pix = nBase % HWSZ;
  int y   = pix >> 7;
  int x0  = pix & 127;

  const float* xb  = x  + (size_t)b * 128 * HWSZ;
  const float* xeb = xe + (size_t)b * 19  * HWSZ;

  for (int i = t; i < CPAD * 54; i += 256) {
    int c = i / 54, e = i % 54;
    int r = e / 18, cc = e % 18;
    int row = y + r - 1, col = x0 + cc - 1;
    float v = 0.f;
    if (c < CIN && (unsigned)row < 128u && (unsigned)col < 128u) {
      int sp = row * IMW + col;
      v = (c < 128) ? xb[(size_t)c * HWSZ + sp]
                    : xeb[(size_t)(xebase + c - 128) * HWSZ + sp];
    }
    patch[((c >> 1) * 56 + e) * 2 + (c & 1)] = v;
  }
  __syncthreads();

  int wave = t >> 5, lane = t & 31;
  int half = lane >> 4, l16 = lane & 15;
  int mW = mWBase + wave * 16;                 // branch-local out-channel base

  const float* abase = wpk + (size_t)(mW + l16) * 2;
  v8f acc = {0.f, 0.f, 0.f, 0.f, 0.f, 0.f, 0.f, 0.f};

  #pragma unroll
  for (int tap = 0; tap < 9; ++tap) {
    int kh = tap / 3, kw = tap % 3;
    int e = kh * 18 + kw + l16;
    #pragma unroll 4
    for (int ci0 = 0; ci0 < CPAD; ci0 += 4) {
      int c2 = (ci0 >> 1) + half;
      v2f a  = *(const v2f*)(abase + (size_t)(tap * (CPAD / 2) + c2) * OC * 2);
      v2f bf = *(const v2f*)(&patch[(c2 * 56 + e) * 2]);
      acc = wmma_f32(a, bf, acc);
    }
  }

  int mOut = mOutBase + wave * 16;
  float* o = out5 + (size_t)b * 640 * HWSZ + (size_t)mOut * HWSZ + pix + l16;
  #pragma unroll
  for (int r = 0; r < 8; ++r) {
    int m = r + 8 * half;
    o[(size_t)m * HWSZ] = (acc[r] + bias[mW + m]) * oscale;
  }
}

// ---------------------------------------------------------------------------
// Kernel 2: fused depthwise 9-tap conv + attention. One block per
// (b, s, head, 16x16 pixel tile). k/v halos (16ch x 18x18) + dep weights in
// LDS. attnout[s][b][ch][pix]
// ---------------------------------------------------------------------------
__global__ __launch_bounds__(256) void dep_attn(
    const float* __restrict__ conv5, const float* __restrict__ depw,
    const float* __restrict__ depb,  float* __restrict__ attnout)
{
  __shared__ float ktile[16 * 324];
  __shared__ float vtile[16 * 324];
  __shared__ float dws[144 * 9];
  __shared__ float dbs[144];

  int t    = threadIdx.x;
  int tile = blockIdx.x & 63;
  int combo = blockIdx.x >> 6;
  int n = combo & 7, s = (combo >> 3) & 1, b = combo >> 4;
  int ty = tile >> 3, tx = tile & 7;

  for (int i = t; i < 1296; i += 256) dws[i] = depw[i];
  if (t < 144) dbs[t] = depb[t];

  int kBase = (s == 0 ? 128 : 384) + n * 16;
  int vBase = (s == 0 ? 256 : 512) + n * 16;
  const float* c5b = conv5 + (size_t)b * 640 * HWSZ;

  for (int i = t; i < 16 * 324; i += 256) {
    int d = i / 324, rem = i % 324, yy = rem / 18, xx = rem % 18;
    int gy = ty * 16 + yy - 1, gx = tx * 16 + xx - 1;
    float kv = 0.f, vv = 0.f;
    if ((unsigned)gy < 128u && (unsigned)gx < 128u) {
      int sp = gy * IMW + gx;
      kv = c5b[(size_t)(kBase + d) * HWSZ + sp];
      vv = c5b[(size_t)(vBase + d) * HWSZ + sp];
    }
    ktile[i] = kv; vtile[i] = vv;
  }
  __syncthreads();

  int yy = t >> 4, xx = t & 15;
  int sp = (ty * 16 + yy) * IMW + (tx * 16 + xx);

  float q[16];
  #pragma unroll
  for (int d = 0; d < 16; ++d) q[d] = c5b[(size_t)(n * 16 + d) * HWSZ + sp];

  float logits[9];
  #pragma unroll
  for (int a = 0; a < 9; ++a) logits[a] = 0.f;

  #pragma unroll
  for (int d = 0; d < 16; ++d) {
    float st[9];
    #pragma unroll
    for (int kh = 0; kh < 3; ++kh)
      #pragma unroll
      for (int kw = 0; kw < 3; ++kw)
        st[kh * 3 + kw] = ktile[d * 324 + (yy + kh) * 18 + (xx + kw)];
    float qd = q[d];
    #pragma unroll
    for (int a = 0; a < 9; ++a) {
      float kt = dbs[d * 9 + a];
      #pragma unroll
      for (int tp = 0; tp < 9; ++tp) kt += st[tp] * dws[(d * 9 + a) * 9 + tp];
      logits[a] += qd * kt;
    }
  }

  float mx = logits[0];
  #pragma unroll
  for (int a = 1; a < 9; ++a) mx = fmaxf(mx, logits[a]);
  float att[9]; float sum = 0.f;
  #pragma unroll
  for (int a = 0; a < 9; ++a) { att[a] = __expf(logits[a] - mx); sum += att[a]; }
  float inv = 1.f / sum;
  #pragma unroll
  for (int a = 0; a < 9; ++a) att[a] *= inv;

  float* ob = attnout + ((size_t)(s * 2 + b) * 128 + n * 16) * HWSZ + sp;
  #pragma unroll
  for (int d = 0; d < 16; ++d) {
    float st[9];
    #pragma unroll
    for (int kh = 0; kh < 3; ++kh)
      #pragma unroll
      for (int kw = 0; kw < 3; ++kw)
        st[kh * 3 + kw] = vtile[d * 324 + (yy + kh) * 18 + (xx + kw)];
    float o = 0.f;
    #pragma unroll
    for (int a = 0; a < 9; ++a) {
      float vt = dbs[d * 9 + a];
      #pragma unroll
      for (int tp = 0; tp < 9; ++tp) vt += st[tp] * dws[(d * 9 + a) * 9 + tp];
      o += att[a] * vt;
    }
    ob[(size_t)d * HWSZ] = o;
  }
}

// ---------------------------------------------------------------------------
// Kernel 3: 1x1 projections as WMMA GEMMs. Block = (s, 16-pixel strip);
// 8 waves = 8 M-tiles sharing an LDS-staged 128ch x 16px activation block.
// ---------------------------------------------------------------------------
__global__ __launch_bounds__(256) void proj(
    const float* __restrict__ attn, const float* __restrict__ wpk,
    const float* __restrict__ bpan, const float* __restrict__ bms,
    float* __restrict__ out)
{
  __shared__ float bs[64 * 16 * 2];   // [cpair][16 pix][2]

  int t = threadIdx.x;
  int nTile = blockIdx.x & 2047;
  int s     = blockIdx.x >> 11;
  int nBase = nTile * 16;
  int b = nBase / HWSZ, pix = nBase % HWSZ;

  const float* src = attn + (size_t)(s * 2 + b) * 128 * HWSZ + pix;
  for (int i = t; i < 2048; i += 256) {
    int c = i >> 4, p = i & 15;
    bs[((c >> 1) * 16 + p) * 2 + (c & 1)] = src[(size_t)c * HWSZ + p];
  }
  __syncthreads();

  int wave = t >> 5, lane = t & 31;
  int half = lane >> 4, l16 = lane & 15;
  const float* bias = s ? bms : bpan;
  const float* wb = wpk + (size_t)s * 16384 + (size_t)(wave * 16 + l16) * 2;

  v8f acc = {0.f, 0.f, 0.f, 0.f, 0.f, 0.f, 0.f, 0.f};
  #pragma unroll
  for (int k0 = 0; k0 < 128; k0 += 4) {
    int c2 = (k0 >> 1) + half;
    v2f a  = *(const v2f*)(wb + (size_t)c2 * 128 * 2);
    v2f bf = *(const v2f*)(&bs[(c2 * 16 + l16) * 2]);
    acc = wmma_f32(a, bf, acc);
  }

  float* o = out + (size_t)s * 2 * 128 * HWSZ + (size_t)b * 128 * HWSZ
               + (size_t)(wave * 16) * HWSZ + pix + l16;
  #pragma unroll
  for (int r = 0; r < 8; ++r) {
    int m = r + 8 * half;
    o[(size_t)m * HWSZ] = acc[r] + bias[wave * 16 + m];
  }
}

// ---------------------------------------------------------------------------
extern "C" void kernel_launch(void* const* d_in, const int* in_sizes, int n_in,
                              void* d_out, int out_size, void* d_ws, size_t ws_size,
                              hipStream_t stream) {
  const float* x    = (const float*)d_in[0];
  const float* ms   = (const float*)d_in[1];
  const float* lpan = (const float*)d_in[2];
  const float* pan  = (const float*)d_in[3];
  const float* sv   = (const float*)d_in[4];
  const float* qw   = (const float*)d_in[5];
  const float* qb   = (const float*)d_in[6];
  const float* kpw  = (const float*)d_in[7];
  const float* kpb  = (const float*)d_in[8];
  const float* vpw  = (const float*)d_in[9];
  const float* vpb  = (const float*)d_in[10];
  const float* kvw  = (const float*)d_in[11];
  const float* kvb  = (const float*)d_in[12];
  const float* depw = (const float*)d_in[13];
  const float* depb = (const float*)d_in[14];
  const float* wpan = (const float*)d_in[15];
  const float* bpan = (const float*)d_in[16];
  const float* wms  = (const float*)d_in[17];
  const float* bms  = (const float*)d_in[18];
  float* out = (float*)d_out;

  char* ws = (char*)d_ws;
  auto align256 = [](size_t v) { return (v + 255) & ~(size_t)255; };
  size_t xeSz   = align256((size_t)2 * 19  * HWSZ * 4);   //  2.4 MB
  size_t c5Sz   = align256((size_t)2 * 640 * HWSZ * 4);   // 83.9 MB
  size_t atSz   = align256((size_t)2 * 2 * 128 * HWSZ * 4); // 33.6 MB
  size_t wqSz   = align256((size_t)9 * 136 * 128 * 4);
  size_t wkpSz  = align256((size_t)9 * 132 * 128 * 4);
  size_t wvpSz  = align256((size_t)9 * 132 * 128 * 4);
  size_t wkvSz  = align256((size_t)9 * 136 * 256 * 4);

  float* xe    = (float*)ws;
  float* conv5 = (float*)(ws + xeSz);
  float* attn  = (float*)(ws + xeSz + c5Sz);
  char*  wbase = ws + xeSz + c5Sz + atSz;
  float* wq  = (float*)wbase;
  float* wkp = (float*)(wbase + wqSz);
  float* wvp = (float*)(wbase + wqSz + wkpSz);
  float* wkv = (float*)(wbase + wqSz + wkpSz + wvpSz);
  float* wpj = (float*)(wbase + wqSz + wkpSz + wvpSz + wkvSz); // 2*128*128

  // --- prep ---
  build_xe<<<128, 256, 0, stream>>>(ms, lpan, pan, sv, xe);
  pack_w<<<(9 * 136 * 128 + 255) / 256, 256, 0, stream>>>(qw,  wq,  136, 136, 128);
  pack_w<<<(9 * 132 * 128 + 255) / 256, 256, 0, stream>>>(kpw, wkp, 129, 132, 128);
  pack_w<<<(9 * 132 * 128 + 255) / 256, 256, 0, stream>>>(vpw, wvp, 131, 132, 128);
  pack_w<<<(9 * 136 * 256 + 255) / 256, 256, 0, stream>>>(kvw, wkv, 136, 136, 256);
  pack_proj<<<(2 * 128 * 128 + 255) / 256, 256, 0, stream>>>(wpan, wms, wpj);

  // --- front convs: q*SCALE, k_pan, v_pan, k_ms, v_ms -> conv5[b][640][pix]
  conv_front_t<136, 136, 128><<<2048, 256, 0, stream>>>(x, xe, 0,  wq,  qb,  0,   0,   0.25f, conv5);
  conv_front_t<129, 132, 128><<<2048, 256, 0, stream>>>(x, xe, 16, wkp, kpb, 128, 0,   1.0f,  conv5);
  conv_front_t<131, 132, 128><<<2048, 256, 0, stream>>>(x, xe, 16, wvp, vpb, 256, 0,   1.0f,  conv5);
  conv_front_t<136, 136, 256><<<2048, 256, 0, stream>>>(x, xe, 8,  wkv, kvb, 384, 0,   1.0f,  conv5);
  conv_front_t<136, 136, 256><<<2048, 256, 0, stream>>>(x, xe, 8,  wkv, kvb, 512, 128, 1.0f,  conv5);

  // --- fused depthwise conv + attention ---
  dep_attn<<<2048, 256, 0, stream>>>(conv5, depw, depb, attn);

  // --- projections -> d_out ---
  proj<<<4096, 256, 0, stream>>>(attn, wpj, bpan, bms, out);
}